// PhysicsInformedLoss_10934986735710
// MI455X (gfx1250) — compile-verified
//
#include <hip/hip_runtime.h>

// ---------------------------------------------------------------------------
// PhysicsInformedLoss for MI455X (gfx1250, wave32).
//   ws layout (floats): [0,N)   accM   signed mass-flux accumulator
//                       [N,4N)  accF   signed momentum-flux accumulator (3/node)
//                       [4N,4N+8) scal 8 global partial sums
// ---------------------------------------------------------------------------

typedef __attribute__((ext_vector_type(2))) float v2f;
typedef __attribute__((ext_vector_type(8))) float v8f;

#define EPS_DIV  1e-8f
#define EPS_REL  1e-6f
#define CLIPMAXF 10.0f

__device__ __forceinline__ void atom_add(float* p, float v) {
  // hardware global_atomic_add_f32 (no CAS loop)
  unsafeAtomicAdd(p, v);
}

// ---------------------------------------------------------------- zero init
__global__ void ws_zero_kernel(float* __restrict__ ws, int n) {
  for (int i = blockIdx.x * blockDim.x + threadIdx.x; i < n;
       i += gridDim.x * blockDim.x)
    ws[i] = 0.0f;
}

// ---------------------------------------------------------------- edge pass
__global__ __launch_bounds__(256) void edge_flux_kernel(
    const long long* __restrict__ ei,   // [2E] row-major: src then dst
    const float* __restrict__ pos,      // [3N]
    const float* __restrict__ rho,      // [N]
    const float* __restrict__ U,        // [3N]
    const float* __restrict__ p,        // [N]
    float* __restrict__ accM,           // [N]
    float* __restrict__ accF,           // [3N]
    int E) {
  int e = blockIdx.x * blockDim.x + threadIdx.x;
  if (e >= E) return;

  // stream-ahead on the only HBM-resident data (emits global_prefetch_b8;
  // speculative, OOB addresses are silently dropped)
  __builtin_prefetch(ei + e + 8192, 0, 1);
  __builtin_prefetch(ei + e + E + 8192, 0, 1);

  int s = (int)ei[e];
  int d = (int)ei[e + E];

  float psx = pos[3 * s + 0], psy = pos[3 * s + 1], psz = pos[3 * s + 2];
  float pdx = pos[3 * d + 0], pdy = pos[3 * d + 1], pdz = pos[3 * d + 2];

  float evx = pdx - psx, evy = pdy - psy, evz = pdz - psz;
  float len = sqrtf(evx * evx + evy * evy + evz * evz) + EPS_DIV;
  float inv = 1.0f / len;
  float eux = evx * inv, euy = evy * inv, euz = evz * inv;
  float area = len * len;

  float rs = rho[s], rd = rho[d];
  float Usx = U[3 * s + 0], Usy = U[3 * s + 1], Usz = U[3 * s + 2];
  float Udx = U[3 * d + 0], Udy = U[3 * d + 1], Udz = U[3 * d + 2];

  float Ufx = 0.5f * (Usx + Udx);
  float Ufy = 0.5f * (Usy + Udy);
  float Ufz = 0.5f * (Usz + Udz);

  float Un = Ufx * eux + Ufy * euy + Ufz * euz;
  float rf = 2.0f * rs * rd / (rs + rd + EPS_DIV);   // harmonic mean
  float fm = rf * Un * area;                         // mass flux
  float pf = 0.5f * (p[s] + p[d]);

  float c   = rf * Un;
  float tfx = c * Ufx + pf * eux;
  float tfy = c * Ufy + pf * euy;
  float tfz = c * Ufz + pf * euz;

  // (out - in) folded into one signed accumulator: +at dst, -at src
  atom_add(accM + d,  fm);
  atom_add(accM + s, -fm);
  atom_add(accF + 3 * d + 0,  tfx);
  atom_add(accF + 3 * d + 1,  tfy);
  atom_add(accF + 3 * d + 2,  tfz);
  atom_add(accF + 3 * s + 0, -tfx);
  atom_add(accF + 3 * s + 1, -tfy);
  atom_add(accF + 3 * s + 2, -tfz);
}

// ---------------------------------------------------------------- node pass
// 8 partial sums: 0:p 1:T 2:Mach 3:U (rel-sq sums), 4:div_mass^2,
// 5:div_mom^2, 6:wall_cnt, 7:wall_sse
__global__ __launch_bounds__(256) void node_stats_kernel(
    const float* __restrict__ pp, const float* __restrict__ pT,
    const float* __restrict__ pM, const float* __restrict__ pU,
    const float* __restrict__ tp, const float* __restrict__ tT,
    const float* __restrict__ tM, const float* __restrict__ tU,
    const float* __restrict__ vol, const float* __restrict__ bt,
    const float* __restrict__ accM, const float* __restrict__ accF,
    float* __restrict__ scal, int N) {
  float a[8] = {0.f, 0.f, 0.f, 0.f, 0.f, 0.f, 0.f, 0.f};

  for (int i = blockIdx.x * blockDim.x + threadIdx.x; i < N;
       i += gridDim.x * blockDim.x) {
    float r;
    r = (pp[i] - tp[i]) / (fabsf(tp[i]) + EPS_REL); a[0] += r * r;
    r = (pT[i] - tT[i]) / (fabsf(tT[i]) + EPS_REL); a[1] += r * r;
    r = (pM[i] - tM[i]) / (fabsf(tM[i]) + EPS_REL); a[2] += r * r;

    float ux = pU[3 * i + 0], uy = pU[3 * i + 1], uz = pU[3 * i + 2];
    float rx = (ux - tU[3 * i + 0]) / (fabsf(tU[3 * i + 0]) + EPS_REL);
    float ry = (uy - tU[3 * i + 1]) / (fabsf(tU[3 * i + 1]) + EPS_REL);
    float rz = (uz - tU[3 * i + 2]) / (fabsf(tU[3 * i + 2]) + EPS_REL);
    a[3] += rx * rx + ry * ry + rz * rz;

    float iv = 1.0f / (vol[i] + EPS_DIV);
    float dm = accM[i] * iv;
    a[4] += dm * dm;
    float fx = accF[3 * i + 0] * iv;
    float fy = accF[3 * i + 1] * iv;
    float fz = accF[3 * i + 2] * iv;
    a[5] += fx * fx + fy * fy + fz * fz;

    float wall = (bt[5 * i + 0] > 0.5f) ? 1.0f : 0.0f;
    a[6] += wall;
    a[7] += wall * (ux * ux + uy * uy + uz * uz);
  }

  // intra-wave butterfly (wave32)
#pragma unroll
  for (int off = 16; off > 0; off >>= 1)
#pragma unroll
    for (int j = 0; j < 8; ++j) a[j] += __shfl_xor(a[j], off, 32);

  __shared__ float S[64];                       // S[wave*8 + q], 8 waves
  int lane = threadIdx.x & 31;
  int wave = threadIdx.x >> 5;
  if (lane == 0) {
#pragma unroll
    for (int j = 0; j < 8; ++j) S[wave * 8 + j] = a[j];
  }
  __syncthreads();

  if (threadIdx.x < 32) {                       // wave 0, EXEC all-ones
    int l  = threadIdx.x;
    int r  = l & 15;                            // A-matrix row M
    int kh = l >> 4;                            // K half: 0 -> K0/K1, 1 -> K2/K3
    int q  = r & 7;                             // quantity
    int g  = r >> 3;                            // wave group (0: w0-3, 1: w4-7)
    int w0 = g * 4 + kh * 2;

#if __has_builtin(__builtin_amdgcn_wmma_f32_16x16x4_f32)
    // Row m=(q,g) of A holds the 4 wave-partials of quantity q in group g;
    // B = ones  =>  D[m,*] = sum_k A[m,k]  (exact f32 row-sums in 1 wmma).
    v2f A; A.x = S[(w0 + 0) * 8 + q]; A.y = S[(w0 + 1) * 8 + q];
    v2f B; B.x = 1.0f;                B.y = 1.0f;
    v8f C = {};
    C = __builtin_amdgcn_wmma_f32_16x16x4_f32(
        /*neg_a=*/false, A, /*neg_b=*/false, B,
        /*c_mod=*/(short)0, C, /*reuse_a=*/false, /*reuse_b=*/false);
    // D vgpr j: lanes 0-15 = rowsum(q=j,g=0), lanes 16-31 = rowsum(q=j,g=1)
#pragma unroll
    for (int j = 0; j < 8; ++j) {
      float t = C[j] + __shfl_xor(C[j], 16, 32);   // g0 + g1 = block total
      if (l == j) atom_add(&scal[j], t);
    }
#else
    if (l < 8) {
      float t = 0.f;
      for (int w = 0; w < 8; ++w) t += S[w * 8 + l];
      atom_add(&scal[l], t);
    }
#endif
  }
}

// ---------------------------------------------------------------- finalize
__global__ void finalize_kernel(const float* __restrict__ scal,
                                float* __restrict__ out, int N) {
  if (threadIdx.x == 0 && blockIdx.x == 0) {
    float n = (float)N;
    float L_data = (scal[0] / n + scal[1] / n + scal[2] / n +
                    scal[3] / (3.0f * n)) * 0.25f;
    float L_mass = fminf(scal[4] / n, CLIPMAXF);
    float L_mom  = fminf(scal[5] / (3.0f * n), CLIPMAXF);
    float cnt = scal[6], sse = scal[7];
    float L_bc = (cnt > 0.0f) ? (sse / (cnt * 3.0f + EPS_DIV)) : 0.0f;
    L_bc = fminf(L_bc, CLIPMAXF);
    out[0] = 1.0f * L_data + 0.05f * L_mass + 0.02f * L_mom + 0.03f * L_bc;
  }
}

// ---------------------------------------------------------------------------
extern "C" void kernel_launch(void* const* d_in, const int* in_sizes, int n_in,
                              void* d_out, int out_size, void* d_ws,
                              size_t ws_size, hipStream_t stream) {
  (void)n_in; (void)out_size; (void)ws_size;

  const float* pred_p    = (const float*)d_in[0];
  const float* pred_T    = (const float*)d_in[1];
  const float* pred_Mach = (const float*)d_in[2];
  const float* pred_U    = (const float*)d_in[3];
  const float* pred_rho  = (const float*)d_in[4];
  const float* target_p  = (const float*)d_in[5];
  const float* target_T  = (const float*)d_in[6];
  const float* target_M  = (const float*)d_in[7];
  const float* target_U  = (const float*)d_in[8];
  const float* node_vol  = (const float*)d_in[9];
  const float* node_pos  = (const float*)d_in[10];
  const float* bnd_type  = (const float*)d_in[11];
  const long long* ei    = (const long long*)d_in[12];   // int64 per reference

  int N = in_sizes[0];
  int E = in_sizes[12] / 2;

  float* ws   = (float*)d_ws;
  float* accM = ws;            // N
  float* accF = ws + N;        // 3N
  float* scal = ws + 4 * N;    // 8
  int nz = 4 * N + 8;

  ws_zero_kernel<<<512, 256, 0, stream>>>(ws, nz);

  edge_flux_kernel<<<(E + 255) / 256, 256, 0, stream>>>(
      ei, node_pos, pred_rho, pred_U, pred_p, accM, accF, E);

  node_stats_kernel<<<384, 256, 0, stream>>>(
      pred_p, pred_T, pred_Mach, pred_U, target_p, target_T, target_M,
      target_U, node_vol, bnd_type, accM, accF, scal, N);

  finalize_kernel<<<1, 32, 0, stream>>>(scal, (float*)d_out, N);
}